// CorefGAT_70437463655035
// MI455X (gfx1250) — compile-verified
//
#include <hip/hip_runtime.h>
#include <hip/hip_bf16.h>
#include <math.h>

typedef __attribute__((ext_vector_type(2))) float v2f;
typedef __attribute__((ext_vector_type(8))) float v8f;

#define IN_DIM 768
#define HID    256
#define HEADS  4
#define NEG_SLOPE 0.2f

// ---------------------------------------------------------------------------
// Tiled f32 WMMA GEMM: C[M,N] = A[M,K] @ B[K,N]   (row-major, K,N mult of 16)
// block = 256 threads (8 waves). Block tile 64(M) x 128(N), K staged 16-wide,
// double-buffered LDS. Wave (wm 0..3, wn 0..1) owns rows wm*16..+16 and cols
// wn*64..+64 as four 16x16 WMMA accumulators.
//
// LDS layouts:
//   As[buf][row*20 + k]          row-major, stride 20 (16B-aligned float4
//                                stores; 16-lane column reads conflict-free)
//   Bs[buf][j*288 + 2*col + p]   K-pair interleaved: pair j = rows 2j,2j+1,
//                                p = row parity. A B-fragment {B[k][n],
//                                B[k+1][n]} is ONE aligned ds_load_b64.
//                                288 mod 64 == 32 -> the lanes16-31 pair row
//                                lands on the other 32 banks (no conflicts).
// ---------------------------------------------------------------------------
#define ASTR  20
#define BSTR2 288

__global__ __launch_bounds__(256)
void gemm_wmma_f32(const float* __restrict__ A, const float* __restrict__ B,
                   float* __restrict__ C, int M, int K, int N) {
    __shared__ float As[2][64 * ASTR];    // 2 x 5120 B
    __shared__ float Bs[2][8 * BSTR2];    // 2 x 9216 B

    const int tid  = threadIdx.x;
    const int bm   = blockIdx.x * 64;
    const int bn   = blockIdx.y * 128;
    const int wave = tid >> 5;
    const int lane = tid & 31;
    const int wm   = wave & 3;        // M-tile within block
    const int wn   = wave >> 2;       // N-half within block
    const int half = lane >> 4;       // 0: lanes 0-15, 1: lanes 16-31
    const int mrow = lane & 15;

    // staging assignments
    const int ar = tid >> 2;          // A row 0..63
    const int ac = (tid & 3) << 2;    // A col 0,4,8,12
    const int bj = tid >> 5;          // B k-pair 0..7
    const int bc = (tid & 31) << 2;   // B col 0..124
    const bool arow_ok = (bm + ar) < M;
    const float* Aptr = A + (size_t)(bm + ar) * K + ac;          // + k0
    const float* Bptr = B + (size_t)(2 * bj) * N + bn + bc;      // + k0*N

    v8f acc[4];
    #pragma unroll
    for (int i = 0; i < 4; ++i) acc[i] = (v8f){0,0,0,0,0,0,0,0};

    float4 av, bv0, bv1;
    auto gload = [&](int k0) {
        av = arow_ok ? *(const float4*)(Aptr + k0)
                     : make_float4(0.f, 0.f, 0.f, 0.f);
        bv0 = *(const float4*)(Bptr + (size_t)k0 * N);
        bv1 = *(const float4*)(Bptr + (size_t)(k0 + 1) * N);
    };
    auto lstore = [&](int buf) {
        *(float4*)(&As[buf][ar * ASTR + ac]) = av;
        float b0[4] = {bv0.x, bv0.y, bv0.z, bv0.w};
        float b1[4] = {bv1.x, bv1.y, bv1.z, bv1.w};
        #pragma unroll
        for (int i = 0; i < 4; ++i)
            *(float2*)(&Bs[buf][bj * BSTR2 + 2 * (bc + i)]) =
                make_float2(b0[i], b1[i]);
    };

    const int nstage = K >> 4;
    gload(0);
    lstore(0);
    __syncthreads();

    for (int s = 0; s < nstage; ++s) {
        const int buf = s & 1;
        const bool more = (s + 1) < nstage;
        if (more) gload((s + 1) << 4);   // issue next-stage HBM loads early

        const float* as_ = As[buf];
        const float* bs_ = Bs[buf];
        #pragma unroll
        for (int kk = 0; kk < 16; kk += 4) {
            // A frag: lanes 0-15 -> K = kk,kk+1 ; lanes 16-31 -> K = kk+2,kk+3
            float2 af = *(const float2*)(&as_[(wm * 16 + mrow) * ASTR + kk + half * 2]);
            v2f a; a.x = af.x; a.y = af.y;
            const int j = (kk >> 1) + half;   // K-pair index for this half-wave
            #pragma unroll
            for (int nt = 0; nt < 4; ++nt) {
                int ncol = wn * 64 + nt * 16 + mrow;
                float2 bf = *(const float2*)(&bs_[j * BSTR2 + 2 * ncol]);
                v2f b; b.x = bf.x; b.y = bf.y;
                acc[nt] = __builtin_amdgcn_wmma_f32_16x16x4_f32(
                    false, a, false, b, (short)0, acc[nt], false, false);
            }
        }

        if (more) {
            __syncthreads();     // readers of buf^1 (stage s-1) are done
            lstore(buf ^ 1);
            __syncthreads();     // stage s+1 tile visible
        }
    }

    // ---- write D: VGPR i -> row (i + half*8), col = mrow within tile ----
    #pragma unroll
    for (int nt = 0; nt < 4; ++nt) {
        int gcol = bn + wn * 64 + nt * 16 + mrow;
        #pragma unroll
        for (int i = 0; i < 8; ++i) {
            int grow = bm + wm * 16 + half * 8 + i;
            if (grow < M) C[(size_t)grow * N + gcol] = acc[nt][i];
        }
    }
}

// ---------------------------------------------------------------------------
// Per-(node,head) attention logits: al_s = <h[n,h,:], a_src[h,:]>, al_d likewise.
// One wave32 per (node, head); shuffle reduction.
// ---------------------------------------------------------------------------
__global__ __launch_bounds__(256)
void attn_logits(const float* __restrict__ h, const float* __restrict__ a_src,
                 const float* __restrict__ a_dst, float* __restrict__ al_s,
                 float* __restrict__ al_d, int n, int heads, int hid) {
    int wid  = (blockIdx.x * 256 + threadIdx.x) >> 5;
    int lane = threadIdx.x & 31;
    if (wid >= n * heads) return;
    int node = wid / heads;
    int hh   = wid - node * heads;
    const float* hp = h + (size_t)node * heads * hid + (size_t)hh * hid;
    const float* as = a_src + hh * hid;
    const float* ad = a_dst + hh * hid;
    float ss = 0.f, sd = 0.f;
    for (int c = lane; c < hid; c += 32) {
        float v = hp[c];
        ss += v * as[c];
        sd += v * ad[c];
    }
    #pragma unroll
    for (int m = 16; m >= 1; m >>= 1) {
        ss += __shfl_xor(ss, m, 32);
        sd += __shfl_xor(sd, m, 32);
    }
    if (lane == 0) { al_s[wid] = ss; al_d[wid] = sd; }
}

// ---------------------------------------------------------------------------
__device__ inline void atomicMaxFloat(float* addr, float val) {
    // Works for any sign mix: positive floats compare as ints, negative as uints.
    if (val >= 0.f) atomicMax((int*)addr, __float_as_int(val));
    else            atomicMin((unsigned int*)addr, __float_as_uint(val));
}

__device__ inline void edge_endpoints(const int* ei, int e, int E, int* s, int* d) {
    if (e < E) { *s = ei[e]; *d = ei[E + e]; }
    else       { *s = *d = e - E; }          // appended self-loop
}

// Pass 1: e = leaky_relu(al_s[src]+al_d[dst]); segment max over dst.
__global__ __launch_bounds__(256)
void edge_logits_max(const int* __restrict__ ei, int E, int n, int heads,
                     const float* __restrict__ al_s, const float* __restrict__ al_d,
                     float* __restrict__ e_buf, float* __restrict__ emax) {
    int e = blockIdx.x * 256 + threadIdx.x;
    if (e >= E + n) return;
    int s, d; edge_endpoints(ei, e, E, &s, &d);
    for (int h = 0; h < heads; ++h) {
        float v = al_s[s * heads + h] + al_d[d * heads + h];
        v = (v > 0.f) ? v : NEG_SLOPE * v;
        e_buf[(size_t)e * heads + h] = v;
        atomicMaxFloat(&emax[d * heads + h], v);
    }
}

// Pass 2: p = exp(e - emax[dst]); segment sum over dst. e_buf overwritten with p.
__global__ __launch_bounds__(256)
void edge_exp_sum(const int* __restrict__ ei, int E, int n, int heads,
                  const float* __restrict__ emax, float* __restrict__ e_buf,
                  float* __restrict__ denom) {
    int e = blockIdx.x * 256 + threadIdx.x;
    if (e >= E + n) return;
    int s, d; edge_endpoints(ei, e, E, &s, &d);
    (void)s;
    for (int h = 0; h < heads; ++h) {
        float p = expf(e_buf[(size_t)e * heads + h] - emax[d * heads + h]);
        e_buf[(size_t)e * heads + h] = p;
        atomicAdd(&denom[d * heads + h], p);
    }
}

// Pass 3: agg[dst] += h[src] * alpha, one block per edge, coalesced channels.
__global__ __launch_bounds__(256)
void edge_aggregate(const int* __restrict__ ei, int E, int n, int heads, int hid,
                    const float* __restrict__ e_p, const float* __restrict__ denom,
                    const float* __restrict__ h, float* __restrict__ agg) {
    int e = blockIdx.x;
    int s, d; edge_endpoints(ei, e, E, &s, &d);
    int F = heads * hid;
    for (int idx = threadIdx.x; idx < F; idx += 256) {
        int hh = idx / hid;
        float alpha = e_p[(size_t)e * heads + hh] / (denom[d * heads + hh] + 1e-16f);
        atomicAdd(&agg[(size_t)d * F + idx], h[(size_t)s * F + idx] * alpha);
    }
}

// ---------------------------------------------------------------------------
__global__ __launch_bounds__(256)
void fill_f32(float* __restrict__ p, float v, size_t count) {
    size_t i = (size_t)blockIdx.x * 256 + threadIdx.x;
    if (i < count) p[i] = v;
}

// x[i] = act(x[i] + b[i % F]);  do_elu => ELU, else identity.
__global__ __launch_bounds__(256)
void bias_act(float* __restrict__ x, const float* __restrict__ b,
              size_t total, int F, int do_elu) {
    size_t i = (size_t)blockIdx.x * 256 + threadIdx.x;
    if (i >= total) return;
    float v = x[i] + b[(int)(i % (size_t)F)];
    if (do_elu) v = (v > 0.f) ? v : (expf(v) - 1.f);
    x[i] = v;
}

// ---------------------------------------------------------------------------
static inline unsigned int div_up(size_t a, size_t b) { return (unsigned int)((a + b - 1) / b); }

extern "C" void kernel_launch(void* const* d_in, const int* in_sizes, int n_in,
                              void* d_out, int out_size, void* d_ws, size_t ws_size,
                              hipStream_t stream) {
    const float* x      = (const float*)d_in[0];
    const int*   ei     = (const int*)  d_in[1];
    const float* W1     = (const float*)d_in[2];
    const float* a_src1 = (const float*)d_in[3];
    const float* a_dst1 = (const float*)d_in[4];
    const float* b1     = (const float*)d_in[5];
    const float* W2     = (const float*)d_in[6];
    const float* a_src2 = (const float*)d_in[7];
    const float* a_dst2 = (const float*)d_in[8];
    const float* b2     = (const float*)d_in[9];
    float* out = (float*)d_out;

    const int N  = in_sizes[0] / IN_DIM;   // 10000
    const int E  = in_sizes[1] / 2;        // 160000
    const int ET = E + N;                  // edges incl. self-loops
    const int F1 = HEADS * HID;            // 1024

    // ---- workspace layout (floats) ----
    float* ws    = (float*)d_ws;
    size_t o = 0;
    float* h1    = ws + o; o += (size_t)N * F1;       // x @ W1
    float* act1  = ws + o; o += (size_t)N * F1;       // agg1 -> ELU(out1)
    float* als1  = ws + o; o += (size_t)N * HEADS;
    float* ald1  = ws + o; o += (size_t)N * HEADS;
    float* e1    = ws + o; o += (size_t)ET * HEADS;
    float* emax1 = ws + o; o += (size_t)N * HEADS;
    float* den1  = ws + o; o += (size_t)N * HEADS;
    float* h2    = ws + o; o += (size_t)N * HID;      // act1 @ W2
    float* als2  = ws + o; o += (size_t)N;
    float* ald2  = ws + o; o += (size_t)N;
    float* e2    = ws + o; o += (size_t)ET;
    float* emax2 = ws + o; o += (size_t)N;
    float* den2  = ws + o; o += (size_t)N;
    (void)ws_size;

    const float NEG_INF = -3.402823466e38f;
    dim3 blk(256);

    // ================= Layer 1 =================
    // h1 = x @ W1   (M=N, K=768, N=1024)
    gemm_wmma_f32<<<dim3(div_up(N, 64), F1 / 128), blk, 0, stream>>>(x, W1, h1, N, IN_DIM, F1);

    attn_logits<<<div_up((size_t)N * HEADS, 8), blk, 0, stream>>>(
        h1, a_src1, a_dst1, als1, ald1, N, HEADS, HID);

    fill_f32<<<div_up((size_t)N * HEADS, 256), blk, 0, stream>>>(emax1, NEG_INF, (size_t)N * HEADS);
    fill_f32<<<div_up((size_t)N * HEADS, 256), blk, 0, stream>>>(den1, 0.f, (size_t)N * HEADS);
    fill_f32<<<div_up((size_t)N * F1, 256), blk, 0, stream>>>(act1, 0.f, (size_t)N * F1);

    edge_logits_max<<<div_up(ET, 256), blk, 0, stream>>>(ei, E, N, HEADS, als1, ald1, e1, emax1);
    edge_exp_sum   <<<div_up(ET, 256), blk, 0, stream>>>(ei, E, N, HEADS, emax1, e1, den1);
    edge_aggregate <<<ET, blk, 0, stream>>>(ei, E, N, HEADS, HID, e1, den1, h1, act1);

    // act1 = ELU(agg1 + b1)
    bias_act<<<div_up((size_t)N * F1, 256), blk, 0, stream>>>(act1, b1, (size_t)N * F1, F1, 1);

    // ================= Layer 2 =================
    // h2 = act1 @ W2   (M=N, K=1024, N=256)
    gemm_wmma_f32<<<dim3(div_up(N, 64), HID / 128), blk, 0, stream>>>(act1, W2, h2, N, F1, HID);

    attn_logits<<<div_up((size_t)N, 8), blk, 0, stream>>>(h2, a_src2, a_dst2, als2, ald2, N, 1, HID);

    fill_f32<<<div_up((size_t)N, 256), blk, 0, stream>>>(emax2, NEG_INF, (size_t)N);
    fill_f32<<<div_up((size_t)N, 256), blk, 0, stream>>>(den2, 0.f, (size_t)N);
    fill_f32<<<div_up((size_t)N * HID, 256), blk, 0, stream>>>(out, 0.f, (size_t)N * HID);

    edge_logits_max<<<div_up(ET, 256), blk, 0, stream>>>(ei, E, N, 1, als2, ald2, e2, emax2);
    edge_exp_sum   <<<div_up(ET, 256), blk, 0, stream>>>(ei, E, N, 1, emax2, e2, den2);
    edge_aggregate <<<ET, blk, 0, stream>>>(ei, E, N, 1, HID, e2, den2, h2, out);

    // out += b2 (no activation on final layer)
    bias_act<<<div_up((size_t)N * HID, 256), blk, 0, stream>>>(out, b2, (size_t)N * HID, HID, 0);
}